// DiffEdgeNodeLayer_80041010528656
// MI455X (gfx1250) — compile-verified
//
#include <hip/hip_runtime.h>
#include <cstdint>

// Problem constants (match reference)
#define B_SZ    2048
#define IN_F    256
#define OUT_F   256

// Tiling: 64 b-rows x 32 o-cols per workgroup; 8 b-rows x 1 o-col per thread.
#define TB      64
#define TO      32
#define THREADS 256           // 8 waves (wave32); wave w owns b-rows [8w, 8w+8)
#define XSTRIDE 256           // x tile row stride (floats): wave-uniform broadcast reads
#define WSTRIDE 260           // weight tile row stride: pad 4 -> <=2-way conflict over 32 o-rows

// CDNA5 async copy: one 16-byte chunk per lane, global -> LDS, tracked by ASYNCcnt.
__device__ __forceinline__ void async_copy16(unsigned lds_byte_addr, const void* gptr) {
    asm volatile("global_load_async_to_lds_b128 %0, %1, off"
                 :: "v"(lds_byte_addr),
                    "v"((unsigned long long)(uintptr_t)gptr)
                 : "memory");
}

// ---------------------------------------------------------------------------
// Kernel 1: hoist the softmaxes out of the O(B*O*I) loop.
//   Ae[o,i] = p_edge, Cn[o,i] = p_noedge, NP[o] = node softmax (2 entries).
//   Fast path: v_rcp_f32 instead of the IEEE division chain.
// ---------------------------------------------------------------------------
__global__ __launch_bounds__(IN_F)
void prep_kernel(const float* __restrict__ pew, const float* __restrict__ pnw,
                 float* __restrict__ Ae, float* __restrict__ Cn,
                 float* __restrict__ NP) {
    const int o = blockIdx.x;
    const int i = threadIdx.x;
    const size_t oi = (size_t)o * IN_F + i;
    const float w0 = pew[oi * 2 + 0];
    const float w1 = pew[oi * 2 + 1];
    const float m  = fmaxf(w0, w1);
    const float e0 = __expf(w0 - m);
    const float e1 = __expf(w1 - m);
    const float inv = __builtin_amdgcn_rcpf(e0 + e1);   // v_rcp_f32 (1 ulp)
    Ae[oi] = e0 * inv;   // p_edge
    Cn[oi] = e1 * inv;   // p_noedge
    if (i == 0) {
        const float n0 = pnw[o * 2 + 0], n1 = pnw[o * 2 + 1];
        const float mm = fmaxf(n0, n1);
        const float f0 = __expf(n0 - mm), f1 = __expf(n1 - mm);
        const float innv = __builtin_amdgcn_rcpf(f0 + f1);
        NP[o * 2 + 0] = f0 * innv;
        NP[o * 2 + 1] = f1 * innv;
    }
}

// ---------------------------------------------------------------------------
// Kernel 2: tiled tropical (min/max-plus) reduction.
//   LDS staged via GLOBAL_LOAD_ASYNC_TO_LDS_B128; ~129 KB of 320 KB/WGP.
//   Each wave: all 32 o-cols x one group of 8 b-rows -> x reads are pure
//   broadcasts, weight reads <=2-way conflicted, output stores coalesced.
// ---------------------------------------------------------------------------
__global__ __launch_bounds__(THREADS)
void eval_kernel(const float* __restrict__ x,  const float* __restrict__ Ae,
                 const float* __restrict__ Cn, const float* __restrict__ NP,
                 float* __restrict__ out) {
    __shared__ float smem[TB * XSTRIDE + 2 * TO * WSTRIDE];
    float* sx = smem;                      // 64 x 256
    float* sa = smem + TB * XSTRIDE;       // 32 x 260 (p_edge)
    float* sc = sa + TO * WSTRIDE;         // 32 x 260 (p_noedge)

    const int t  = threadIdx.x;
    const int b0 = blockIdx.x * TB;
    const int o0 = blockIdx.y * TO;

    // ---- stage tiles: async global -> LDS (ASYNCcnt tracked) ----
    // x tile: TB*IN_F floats = 4096 16B-chunks -> 16 per thread
    #pragma unroll
    for (int j = 0; j < (TB * IN_F / 4) / THREADS; ++j) {
        const int idx = j * THREADS + t;
        const int row = idx >> 6;            // IN_F/4 = 64 chunks per row
        const int col = idx & 63;
        async_copy16((unsigned)(uintptr_t)&sx[row * XSTRIDE + col * 4],
                     x + (size_t)(b0 + row) * IN_F + col * 4);
    }
    // weight tiles: TO*IN_F floats each = 2048 chunks -> 8 per thread per array
    #pragma unroll
    for (int j = 0; j < (TO * IN_F / 4) / THREADS; ++j) {
        const int idx = j * THREADS + t;
        const int row = idx >> 6;
        const int col = idx & 63;
        async_copy16((unsigned)(uintptr_t)&sa[row * WSTRIDE + col * 4],
                     Ae + (size_t)(o0 + row) * IN_F + col * 4);
        async_copy16((unsigned)(uintptr_t)&sc[row * WSTRIDE + col * 4],
                     Cn + (size_t)(o0 + row) * IN_F + col * 4);
    }
    asm volatile("s_wait_asynccnt 0x0" ::: "memory");   // own async copies landed
    __syncthreads();                                    // publish to all 8 waves

    // ---- register tile: 8 b-rows x 1 o-col per thread ----
    const int ol = t & (TO - 1);        // 0..31: lane -> o column (full wave coverage)
    const int wv = t >> 5;              // 0..7:  wave -> b-row group (wave-uniform)
    const float* ar = sa + ol * WSTRIDE;
    const float* cr = sc + ol * WSTRIDE;
    const float* xr = sx + (wv * 8) * XSTRIDE;

    float vmin[8], vmax[8];
    #pragma unroll
    for (int k = 0; k < 8; ++k) { vmin[k] = __builtin_inff(); vmax[k] = -__builtin_inff(); }

    for (int i = 0; i < IN_F; i += 4) {
        const float4 av = *(const float4*)(ar + i);
        const float4 cv = *(const float4*)(cr + i);
        const float4 nv = make_float4(-cv.x, -cv.y, -cv.z, -cv.w);
        #pragma unroll
        for (int k = 0; k < 8; ++k) {
            const float4 xv = *(const float4*)(xr + k * XSTRIDE + i);   // broadcast
            vmin[k] = fminf(vmin[k], fmaf(xv.x, av.x, cv.x));
            vmin[k] = fminf(vmin[k], fmaf(xv.y, av.y, cv.y));
            vmin[k] = fminf(vmin[k], fmaf(xv.z, av.z, cv.z));
            vmin[k] = fminf(vmin[k], fmaf(xv.w, av.w, cv.w));
            vmax[k] = fmaxf(vmax[k], fmaf(xv.x, av.x, nv.x));
            vmax[k] = fmaxf(vmax[k], fmaf(xv.y, av.y, nv.y));
            vmax[k] = fmaxf(vmax[k], fmaf(xv.z, av.z, nv.z));
            vmax[k] = fmaxf(vmax[k], fmaf(xv.w, av.w, nv.w));
        }
    }

    const int oo = o0 + ol;
    const float np0 = NP[oo * 2 + 0];
    const float np1 = NP[oo * 2 + 1];
    #pragma unroll
    for (int k = 0; k < 8; ++k) {
        const int bb = b0 + wv * 8 + k;
        // lanes 0..31 -> consecutive oo: fully coalesced 128B store per wave
        out[(size_t)bb * OUT_F + oo] = fmaf(vmin[k], np0, vmax[k] * np1);
    }
}

// ---------------------------------------------------------------------------
extern "C" void kernel_launch(void* const* d_in, const int* in_sizes, int n_in,
                              void* d_out, int out_size, void* d_ws, size_t ws_size,
                              hipStream_t stream) {
    const float* x   = (const float*)d_in[0];   // (B, IN_F)
    const float* pew = (const float*)d_in[1];   // (OUT_F, IN_F, 2)
    const float* pnw = (const float*)d_in[2];   // (OUT_F, 2)
    float* out = (float*)d_out;                 // (B, OUT_F)

    // Workspace: p_edge (64K floats) | p_noedge (64K floats) | node probs (512 floats)
    float* Ae = (float*)d_ws;
    float* Cn = Ae + (size_t)OUT_F * IN_F;
    float* NP = Cn + (size_t)OUT_F * IN_F;

    prep_kernel<<<OUT_F, IN_F, 0, stream>>>(pew, pnw, Ae, Cn, NP);
    eval_kernel<<<dim3(B_SZ / TB, OUT_F / TO), THREADS, 0, stream>>>(x, Ae, Cn, NP, out);
}